// Bidirectional_Interaction_Type_Attention_85057532330131
// MI455X (gfx1250) — compile-verified
//
#include <hip/hip_runtime.h>
#include <math.h>

// Problem dims (fixed by the reference)
#define BB 16
#define RR 1024
#define FF 128
#define DD 128
#define KK 7
#define HH 512
#define NEGBIG (-1e9f)

typedef __attribute__((ext_vector_type(2))) float v2f;
typedef __attribute__((ext_vector_type(8))) float v8f;
typedef __attribute__((ext_vector_type(4))) unsigned int u32x4;
typedef __attribute__((ext_vector_type(8))) int i32x8;
typedef __attribute__((ext_vector_type(4))) int i32x4;

// -------------------------------------------------------------------------
// LDS byte offset of a __shared__ object: generic LDS addresses carry the
// byte offset in addr[31:0] (aperture bits live in addr[63:32]).
__device__ __forceinline__ unsigned int lds_addr_of(const void* p) {
    return (unsigned int)(uintptr_t)p;
}

// Tensor Data Mover: 2-D f32 tile load global -> LDS (D# per ISA 08 §8).
//   tile_w/tile_h   : tile dims in elements
//   row_stride      : tensor dim0 stride in elements
//   pad_int/pad_amt : LDS pad codes (interval = 2<<code dwords, amount = code+1 dwords)
__device__ __forceinline__ void tdm_load_2d(const void* gptr, unsigned int lds_off,
                                            unsigned int tile_w, unsigned int tile_h,
                                            unsigned int tensor_w, unsigned int tensor_h,
                                            unsigned int row_stride,
                                            unsigned int pad_int, unsigned int pad_amt) {
    unsigned long long ga = (unsigned long long)(uintptr_t)gptr;
    u32x4 g0;
    g0.x = 1u;                                   // count=1 (valid user descriptor)
    g0.y = lds_off;                              // lds_addr [63:32]
    g0.z = (unsigned int)ga;                     // global_addr [95:64]
    g0.w = (unsigned int)((ga >> 32) & 0x1FFFFFFu) | (2u << 30);  // addr[56:32] | type=2
    i32x8 g1;
    g1[0] = (int)((2u << 16)                     // data_size = 4B
                | (1u << 20)                     // pad_enable
                | (pad_int << 22) | (pad_amt << 25));
    g1[1] = (int)((tensor_w & 0xFFFFu) << 16);                         // dim0 lo16
    g1[2] = (int)(((tensor_w >> 16) & 0xFFFFu) | ((tensor_h & 0xFFFFu) << 16));
    g1[3] = (int)(((tensor_h >> 16) & 0xFFFFu) | ((tile_w & 0xFFFFu) << 16));
    g1[4] = (int)(tile_h & 0xFFFFu);             // tile_dim1 | tile_dim2=0
    g1[5] = (int)row_stride;                     // tensor_dim0_stride lo32
    g1[6] = 0;
    g1[7] = 0;
    i32x4 gz4 = {0, 0, 0, 0};                    // groups 2/3: unused for 2-D tile
    i32x8 gz8 = {0, 0, 0, 0, 0, 0, 0, 0};
    // amdgpu-toolchain (clang-23) 6-arg form
    __builtin_amdgcn_tensor_load_to_lds(g0, g1, gz4, gz4, gz8, 0);
}

// -------------------------------------------------------------------------
// softplus of the K type weights
__global__ void k_softplus(const float* __restrict__ tw, float* __restrict__ wsp) {
    int k = threadIdx.x;
    if (k < KK) {
        float x = tw[k];
        wsp[k] = fmaxf(x, 0.f) + log1pf(__expf(-fabsf(x)));
    }
}

// -------------------------------------------------------------------------
// Per-(b,r): K-softmax of logits, weighted sum -> S, masked softmax over F,
// prot_score row sums, per-(b,r) type-strength partials.
__global__ __launch_bounds__(128)
void k_S(const float* __restrict__ logits, const float* __restrict__ fg_mask,
         const float* __restrict__ prot_mask, const float* __restrict__ wsp,
         float* __restrict__ Sbuf, float* __restrict__ attnR,
         float* __restrict__ prot_score, float* __restrict__ typeP) {
    int b = blockIdx.x >> 10;      // / RR
    int r = blockIdx.x & (RR - 1);
    int f = threadIdx.x;

    __shared__ float sred[128];
    __shared__ float tsh[128][8];

    const float* lp = logits + (((size_t)(b * RR + r)) * FF + f) * KK;
    float l[KK];
#pragma unroll
    for (int k = 0; k < KK; ++k) l[k] = lp[k];
    float mk = l[0];
#pragma unroll
    for (int k = 1; k < KK; ++k) mk = fmaxf(mk, l[k]);
    float e[KK], se = 0.f;
#pragma unroll
    for (int k = 0; k < KK; ++k) { e[k] = __expf(l[k] - mk); se += e[k]; }
    float inv = 1.f / se;
    float Sraw = 0.f;
#pragma unroll
    for (int k = 0; k < KK; ++k) {
        float wt = e[k] * inv * wsp[k];   // p * softplus(w): pre-mask "weighted"
        tsh[f][k] = wt;
        Sraw += wt;
    }
    float pm = prot_mask[b * RR + r];
    float fm = fg_mask[b * FF + f];
    float Sv = Sraw * pm * fm;
    Sbuf[((size_t)(b * RR + r)) * FF + f] = Sv;

    float mrow = Sv + (1.f - fm) * NEGBIG;

    sred[f] = mrow; __syncthreads();
    for (int s = 64; s > 0; s >>= 1) { if (f < s) sred[f] = fmaxf(sred[f], sred[f + s]); __syncthreads(); }
    float rowmax = sred[0]; __syncthreads();

    float ex = __expf(mrow - rowmax);
    sred[f] = ex; __syncthreads();
    for (int s = 64; s > 0; s >>= 1) { if (f < s) sred[f] += sred[f + s]; __syncthreads(); }
    float rowsum = sred[0]; __syncthreads();
    attnR[((size_t)(b * RR + r)) * FF + f] = ex / rowsum;

    sred[f] = Sv; __syncthreads();
    for (int s = 64; s > 0; s >>= 1) { if (f < s) sred[f] += sred[f + s]; __syncthreads(); }
    if (f == 0) prot_score[b * RR + r] = sred[0] * pm;
    __syncthreads();

    if (f < KK) {
        float s = 0.f;
        for (int i = 0; i < 128; ++i) s += tsh[i][f];
        typeP[((size_t)(b * RR + r)) * 8 + f] = s;
    }
}

// -------------------------------------------------------------------------
// Per-(b,f): masked softmax over R -> attnT stored [B,F,R]; lig_score.
__global__ __launch_bounds__(256)
void k_softmaxR(const float* __restrict__ Sbuf, const float* __restrict__ prot_mask,
                const float* __restrict__ fg_mask, float* __restrict__ attnT,
                float* __restrict__ lig_score) {
    int b = blockIdx.x >> 7;        // / FF
    int f = blockIdx.x & (FF - 1);
    int tid = threadIdx.x;
    __shared__ float sred[256];

    float mv[4], mloc = -INFINITY, ssum = 0.f;
#pragma unroll
    for (int j = 0; j < 4; ++j) {
        int r = tid + j * 256;
        float s = Sbuf[((size_t)(b * RR + r)) * FF + f];
        float pm = prot_mask[b * RR + r];
        mv[j] = s + (1.f - pm) * NEGBIG;
        mloc = fmaxf(mloc, mv[j]);
        ssum += s;
    }
    sred[tid] = mloc; __syncthreads();
    for (int s = 128; s > 0; s >>= 1) { if (tid < s) sred[tid] = fmaxf(sred[tid], sred[tid + s]); __syncthreads(); }
    float mx = sred[0]; __syncthreads();

    float ev[4], eloc = 0.f;
#pragma unroll
    for (int j = 0; j < 4; ++j) { ev[j] = __expf(mv[j] - mx); eloc += ev[j]; }
    sred[tid] = eloc; __syncthreads();
    for (int s = 128; s > 0; s >>= 1) { if (tid < s) sred[tid] += sred[tid + s]; __syncthreads(); }
    float invs = 1.f / sred[0]; __syncthreads();
#pragma unroll
    for (int j = 0; j < 4; ++j)
        attnT[((size_t)(b * FF + f)) * RR + tid + j * 256] = ev[j] * invs;

    sred[tid] = ssum; __syncthreads();
    for (int s = 128; s > 0; s >>= 1) { if (tid < s) sred[tid] += sred[tid + s]; __syncthreads(); }
    if (tid == 0) lig_score[b * FF + f] = sred[0] * fg_mask[b * FF + f];
}

// -------------------------------------------------------------------------
// Per-batch normalizers + type_strength
__global__ __launch_bounds__(256)
void k_reduce(const float* __restrict__ prot_score, const float* __restrict__ lig_score,
              const float* __restrict__ typeP, float* __restrict__ invP,
              float* __restrict__ invL, float* __restrict__ tstr) {
    int b = blockIdx.x, tid = threadIdx.x;
    __shared__ float sred[256];

    float s = 0.f;
    for (int r = tid; r < RR; r += 256) s += prot_score[b * RR + r];
    sred[tid] = s; __syncthreads();
    for (int st = 128; st > 0; st >>= 1) { if (tid < st) sred[tid] += sred[tid + st]; __syncthreads(); }
    if (tid == 0) invP[b] = 1.f / (sred[0] + 1e-8f);
    __syncthreads();

    s = (tid < FF) ? lig_score[b * FF + tid] : 0.f;
    sred[tid] = s; __syncthreads();
    for (int st = 128; st > 0; st >>= 1) { if (tid < st) sred[tid] += sred[tid + st]; __syncthreads(); }
    if (tid == 0) invL[b] = 1.f / (sred[0] + 1e-8f);
    __syncthreads();

    for (int k = 0; k < KK; ++k) {
        s = 0.f;
        for (int r = tid; r < RR; r += 256) s += typeP[((size_t)(b * RR + r)) * 8 + k];
        sred[tid] = s; __syncthreads();
        for (int st = 128; st > 0; st >>= 1) { if (tid < st) sred[tid] += sred[tid + st]; __syncthreads(); }
        if (tid == 0) tstr[b * 8 + k] = sred[0];
        __syncthreads();
    }
}

// -------------------------------------------------------------------------
// WMMA fused: ligand_context = attnR x ligand_emb (A and B tiles TDM-staged
// into LDS); enr = ctx @ Wr^T + prot_emb + br; score-weighted pool partials.
// One block = one 16-row tile of R.  At stride 132, Bt stride 136 (TDM pad).
__global__ __launch_bounds__(256)
void k_gemm_pool_prot(const float* __restrict__ attnR, const float* __restrict__ lig_emb,
                      const float* __restrict__ prot_emb, const float* __restrict__ Wr,
                      const float* __restrict__ br, const float* __restrict__ prot_score,
                      const float* __restrict__ invP, float* __restrict__ poolP) {
    __shared__ float At[16][132];
    __shared__ float Bt[32][136];
    __shared__ float Ct[16][132];
    int b = blockIdx.x >> 6;        // / 64 tiles
    int mt = blockIdx.x & 63;
    int mBase = mt << 4;
    int tid = threadIdx.x;
    int wave = tid >> 5, lane = tid & 31, half = lane >> 4, lid = lane & 15;
    int nBase = wave << 4;

    // A tile 16x128 (pad 4 dwords per 128 -> LDS stride 132)
    if (tid < 32) {
        tdm_load_2d(attnR + ((size_t)(b * RR + mBase)) * FF, lds_addr_of(&At[0][0]),
                    128, 16, 128, 16, FF, /*pad_int(128dw)*/6, /*pad 4dw*/3);
    }

    v8f acc = {0, 0, 0, 0, 0, 0, 0, 0};
    for (int ch = 0; ch < FF; ch += 32) {
        // B chunk: 32 rows of ligand_emb (pad 8 dwords -> stride 136)
        if (tid < 32) {
            tdm_load_2d(lig_emb + ((size_t)(b * FF + ch)) * DD, lds_addr_of(&Bt[0][0]),
                        128, 32, 128, 32, DD, 6, /*pad 8dw*/7);
            __builtin_amdgcn_s_wait_tensorcnt(0);
        }
        __syncthreads();
        for (int kk = 0; kk < 32; kk += 4) {
            v2f a, bm;
            int r0 = kk + 2 * half;              // lanes 0-15: K=0/1; 16-31: K=2/3
            a.x = At[lid][ch + r0];
            a.y = At[lid][ch + r0 + 1];
            bm.x = Bt[r0][nBase + lid];
            bm.y = Bt[r0 + 1][nBase + lid];
            acc = __builtin_amdgcn_wmma_f32_16x16x4_f32(false, a, false, bm,
                                                        (short)0, acc, false, false);
        }
        __syncthreads();                         // Bt consumed before next DMA
    }
#pragma unroll
    for (int v = 0; v < 8; ++v) Ct[v + 8 * half][nBase + lid] = acc[v];
    __syncthreads();

    // Stage 2: enr = ctx @ Wr^T  (B[e][d] = Wr[d][e]; Wr stays L2-resident)
    v8f acc2 = {0, 0, 0, 0, 0, 0, 0, 0};
    for (int kk = 0; kk < DD; kk += 4) {
        v2f a, bm;
        int c0 = kk + 2 * half;
        a.x = Ct[lid][c0];
        a.y = Ct[lid][c0 + 1];
        const float* wp = Wr + (size_t)(nBase + lid) * DD + c0;
        bm.x = wp[0];
        bm.y = wp[1];
        acc2 = __builtin_amdgcn_wmma_f32_16x16x4_f32(false, a, false, bm,
                                                     (short)0, acc2, false, false);
    }

    float inv = invP[b];
    int dcol = nBase + lid;
    float s8 = 0.f;
#pragma unroll
    for (int v = 0; v < 8; ++v) {
        int m = mBase + v + 8 * half;
        float val = acc2[v] + prot_emb[((size_t)(b * RR + m)) * DD + dcol] + br[dcol];
        s8 += prot_score[b * RR + m] * inv * val;
    }
    poolP[(((size_t)blockIdx.x) * 2 + half) * 128 + dcol] = s8;
}

// -------------------------------------------------------------------------
// Ligand side: prot_context = attnT x protein_emb (K=R, TDM-staged chunks),
// enr = ctx @ Wf^T + lig_emb + bf; lig pool partials.
__global__ __launch_bounds__(256)
void k_gemm_pool_lig(const float* __restrict__ attnT, const float* __restrict__ prot_emb,
                     const float* __restrict__ lig_emb, const float* __restrict__ Wf,
                     const float* __restrict__ bf, const float* __restrict__ lig_score,
                     const float* __restrict__ invL, float* __restrict__ poolL) {
    __shared__ float At[16][132];
    __shared__ float Bt[32][136];
    __shared__ float Ct[16][132];
    int b = blockIdx.x >> 3;        // / 8 tiles
    int mt = blockIdx.x & 7;
    int mBase = mt << 4;
    int tid = threadIdx.x;
    int wave = tid >> 5, lane = tid & 31, half = lane >> 4, lid = lane & 15;
    int nBase = wave << 4;

    v8f acc = {0, 0, 0, 0, 0, 0, 0, 0};
    for (int ch128 = 0; ch128 < RR; ch128 += 128) {
        // A tile 16x128 slice of attnT (row stride R), refreshed per 128-K chunk
        if (tid < 32) {
            tdm_load_2d(attnT + ((size_t)(b * FF + mBase)) * RR + ch128,
                        lds_addr_of(&At[0][0]), 128, 16, 128, 16, RR, 6, 3);
        }
        for (int sub = 0; sub < 128; sub += 32) {
            if (tid < 32) {
                tdm_load_2d(prot_emb + ((size_t)(b * RR + ch128 + sub)) * DD,
                            lds_addr_of(&Bt[0][0]), 128, 32, 128, 32, DD, 6, 7);
                __builtin_amdgcn_s_wait_tensorcnt(0);   // covers At + Bt (in-order)
            }
            __syncthreads();
            for (int kk = 0; kk < 32; kk += 4) {
                v2f a, bm;
                int r0 = kk + 2 * half;
                a.x = At[lid][sub + r0];
                a.y = At[lid][sub + r0 + 1];
                bm.x = Bt[r0][nBase + lid];
                bm.y = Bt[r0 + 1][nBase + lid];
                acc = __builtin_amdgcn_wmma_f32_16x16x4_f32(false, a, false, bm,
                                                            (short)0, acc, false, false);
            }
            __syncthreads();
        }
    }
#pragma unroll
    for (int v = 0; v < 8; ++v) Ct[v + 8 * half][nBase + lid] = acc[v];
    __syncthreads();

    v8f acc2 = {0, 0, 0, 0, 0, 0, 0, 0};
    for (int kk = 0; kk < DD; kk += 4) {
        v2f a, bm;
        int c0 = kk + 2 * half;
        a.x = Ct[lid][c0];
        a.y = Ct[lid][c0 + 1];
        const float* wp = Wf + (size_t)(nBase + lid) * DD + c0;
        bm.x = wp[0];
        bm.y = wp[1];
        acc2 = __builtin_amdgcn_wmma_f32_16x16x4_f32(false, a, false, bm,
                                                     (short)0, acc2, false, false);
    }

    float inv = invL[b];
    int dcol = nBase + lid;
    float s8 = 0.f;
#pragma unroll
    for (int v = 0; v < 8; ++v) {
        int m = mBase + v + 8 * half;
        float val = acc2[v] + lig_emb[((size_t)(b * FF + m)) * DD + dcol] + bf[dcol];
        s8 += lig_score[b * FF + m] * inv * val;
    }
    poolL[(((size_t)blockIdx.x) * 2 + half) * 128 + dcol] = s8;
}

// -------------------------------------------------------------------------
// Reduce pool partials, build final_feat, normalize (output 2), MLP (output 1).
__global__ __launch_bounds__(256)
void k_final(const float* __restrict__ poolP, const float* __restrict__ poolL,
             const float* __restrict__ tstr, const float* __restrict__ W1,
             const float* __restrict__ b1, const float* __restrict__ W2,
             const float* __restrict__ b2, const float* __restrict__ W3,
             const float* __restrict__ b3, float* __restrict__ out) {
    const int IN = 2 * DD + KK;     // 263
    int b = blockIdx.x, tid = threadIdx.x;
    __shared__ float ff[264];
    __shared__ float h1[HH];
    __shared__ float h2[HH / 2];
    __shared__ float sred[256];

    if (tid < 128) {
        float s = 0.f;
        for (int j = 0; j < 128; ++j) s += poolP[((size_t)(b * 128 + j)) * 128 + tid];
        ff[tid] = s;
        float s2 = 0.f;
        for (int j = 0; j < 16; ++j) s2 += poolL[((size_t)(b * 16 + j)) * 128 + tid];
        ff[128 + tid] = s2;
    }
    if (tid < KK) ff[256 + tid] = tstr[b * 8 + tid];
    __syncthreads();

    float sq = ff[tid] * ff[tid];
    if (tid < KK) sq += ff[256 + tid] * ff[256 + tid];
    sred[tid] = sq; __syncthreads();
    for (int s = 128; s > 0; s >>= 1) { if (tid < s) sred[tid] += sred[tid + s]; __syncthreads(); }
    float invn = 1.f / fmaxf(sqrtf(sred[0]), 1e-12f);
    __syncthreads();

    float* ffn = out + 16 + (size_t)b * IN;
    ffn[tid] = ff[tid] * invn;
    if (tid < KK) ffn[256 + tid] = ff[256 + tid] * invn;

    for (int row = tid; row < HH; row += 256) {
        float acc = b1[row];
        const float* w = W1 + (size_t)row * IN;
        for (int i = 0; i < IN; ++i) acc += w[i] * ff[i];
        h1[row] = fmaxf(acc, 0.f);
    }
    __syncthreads();
    {
        float acc = b2[tid];
        const float* w = W2 + (size_t)tid * HH;
        for (int i = 0; i < HH; ++i) acc += w[i] * h1[i];
        h2[tid] = fmaxf(acc, 0.f);
    }
    __syncthreads();
    sred[tid] = h2[tid] * W3[tid]; __syncthreads();
    for (int s = 128; s > 0; s >>= 1) { if (tid < s) sred[tid] += sred[tid + s]; __syncthreads(); }
    if (tid == 0) out[b] = sred[0] + b3[0];
}

// -------------------------------------------------------------------------
extern "C" void kernel_launch(void* const* d_in, const int* in_sizes, int n_in,
                              void* d_out, int out_size, void* d_ws, size_t ws_size,
                              hipStream_t stream) {
    (void)in_sizes; (void)n_in; (void)out_size; (void)ws_size;
    const float* lig_emb    = (const float*)d_in[0];
    const float* prot_emb   = (const float*)d_in[1];
    const float* logits     = (const float*)d_in[2];
    const float* fg_mask    = (const float*)d_in[3];
    const float* prot_mask  = (const float*)d_in[4];
    const float* type_w     = (const float*)d_in[5];
    const float* Wr         = (const float*)d_in[6];
    const float* br         = (const float*)d_in[7];
    const float* Wf         = (const float*)d_in[8];
    const float* bf         = (const float*)d_in[9];
    const float* W1         = (const float*)d_in[10];
    const float* b1         = (const float*)d_in[11];
    const float* W2         = (const float*)d_in[12];
    const float* b2         = (const float*)d_in[13];
    const float* W3         = (const float*)d_in[14];
    const float* b3         = (const float*)d_in[15];
    float* out = (float*)d_out;
    float* ws  = (float*)d_ws;

    size_t off = 0;
    auto take = [&](size_t n) { size_t o = off; off += (n + 63) & ~(size_t)63; return o; };
    float* wsp        = ws + take(8);
    float* Sbuf       = ws + take((size_t)BB * RR * FF);
    float* attnR      = ws + take((size_t)BB * RR * FF);
    float* attnT      = ws + take((size_t)BB * FF * RR);
    float* prot_score = ws + take(BB * RR);
    float* lig_score  = ws + take(BB * FF);
    float* typeP      = ws + take((size_t)BB * RR * 8);
    float* invP       = ws + take(BB);
    float* invL       = ws + take(BB);
    float* tstr       = ws + take(BB * 8);
    float* poolP      = ws + take((size_t)BB * 64 * 2 * 128);
    float* poolL      = ws + take((size_t)BB * 8 * 2 * 128);
    // total ~27 MB of workspace

    k_softplus<<<1, 32, 0, stream>>>(type_w, wsp);
    k_S<<<BB * RR, 128, 0, stream>>>(logits, fg_mask, prot_mask, wsp,
                                     Sbuf, attnR, prot_score, typeP);
    k_softmaxR<<<BB * FF, 256, 0, stream>>>(Sbuf, prot_mask, fg_mask, attnT, lig_score);
    k_reduce<<<BB, 256, 0, stream>>>(prot_score, lig_score, typeP, invP, invL, tstr);
    k_gemm_pool_prot<<<BB * 64, 256, 0, stream>>>(attnR, lig_emb, prot_emb, Wr, br,
                                                  prot_score, invP, poolP);
    k_gemm_pool_lig<<<BB * 8, 256, 0, stream>>>(attnT, prot_emb, lig_emb, Wf, bf,
                                                lig_score, invL, poolL);
    k_final<<<BB, 256, 0, stream>>>(poolP, poolL, tstr, W1, b1, W2, b2, W3, b3, out);
}